// FraudDetectionGNN_89369679495194
// MI455X (gfx1250) — compile-verified
//
#include <hip/hip_runtime.h>
#include <hip/hip_bf16.h>

typedef __attribute__((ext_vector_type(2))) float v2f;
typedef __attribute__((ext_vector_type(8))) float v8f;

#define GNN_N_NODES 50000
#define GNN_N_EDGES 800000
#define GNN_HID     128
#define GNN_LEAKY   0.01f
#define GNN_PF_DIST 4096   // edge-prefetch distance for the gather stream

// ---------------- degree accumulation ----------------
__global__ void gnn_degree_kernel(const int* __restrict__ src, const int* __restrict__ dst,
                                  float* __restrict__ deg_out, float* __restrict__ deg_in, int E) {
    int e = blockIdx.x * blockDim.x + threadIdx.x;
    if (e >= E) return;
    atomicAdd(&deg_out[src[e]], 1.0f);
    atomicAdd(&deg_in[dst[e]], 1.0f);
}

// deg -> rsqrt(max(deg,1)) in place
__global__ void gnn_norm_kernel(float* __restrict__ d, int n) {
    int i = blockIdx.x * blockDim.x + threadIdx.x;
    if (i < n) d[i] = rsqrtf(fmaxf(d[i], 1.0f));
}

// h[i] = x[i] * norm_out[row]   (elementwise, row = i / 128)
__global__ void gnn_scale_kernel(const float* __restrict__ x, const float* __restrict__ nrm,
                                 float* __restrict__ h, int total) {
    int i = blockIdx.x * blockDim.x + threadIdx.x;
    if (i >= total) return;
    h[i] = x[i] * nrm[i >> 7];
}

// warp-per-edge scatter-add: agg[dst] += h[src]  (float4 per lane, 128 floats per edge)
// Lane 0 prefetches the gather row GNN_PF_DIST edges ahead to deepen the L2/HBM pipeline.
__global__ void gnn_scatter_kernel(const float4* __restrict__ h, const int* __restrict__ src,
                                   const int* __restrict__ dst, float* __restrict__ agg, int E) {
    int w = (blockIdx.x * blockDim.x + threadIdx.x) >> 5;
    int lane = threadIdx.x & 31;
    if (w >= E) return;
    if (lane == 0 && w + GNN_PF_DIST < E) {
        int sp = src[w + GNN_PF_DIST];
        __builtin_prefetch(h + (size_t)sp * 32, 0, 3);   // global_prefetch_b8
    }
    int s = src[w];
    int d = dst[w];
    float4 v = h[(size_t)s * 32 + lane];
    float* dr = agg + (size_t)d * GNN_HID + lane * 4;
    atomicAdd(dr + 0, v.x);
    atomicAdd(dr + 1, v.y);
    atomicAdd(dr + 2, v.z);
    atomicAdd(dr + 3, v.w);
}

// ---------------- fused WMMA fp32 GEMM + LayerNorm + LeakyReLU + norm_out prescale -------
// h_out[16 rows] = leakyrelu(layernorm((agg*norm_in) @ W + b)) * norm_out
// grid = N_NODES/16 blocks of 256 threads (8 waves); wave w computes N-tile w (16 cols).
// The block owns complete 128-wide rows, so LN fuses via an LDS round-trip.
__global__ __launch_bounds__(256)
void gnn_gconv_wmma_ln_kernel(const float* __restrict__ agg, const float* __restrict__ norm_in,
                              const float* __restrict__ W, const float* __restrict__ bias,
                              const float* __restrict__ g, const float* __restrict__ be,
                              const float* __restrict__ norm_out, float* __restrict__ h_out) {
    __shared__ float sA[16 * 132];   // 16x128 A tile, padded stride 132 (bank-conflict free)
    __shared__ float sY[16 * 132];   // 16x128 GEMM result tile for the LN pass
    const int tid = threadIdx.x;
    const int m0  = blockIdx.x * 16;

    // ---- stage raw A tile into LDS with async global->LDS DMA (B128 per transfer) ----
    for (int t = tid; t < 16 * 32; t += 256) {
        int r  = t >> 5;
        int c4 = t & 31;
        const float* gsrc = agg + (size_t)(m0 + r) * GNN_HID + c4 * 4;
        unsigned lds_off = (unsigned)(uintptr_t)(&sA[r * 132 + c4 * 4]);
        asm volatile("global_load_async_to_lds_b128 %0, %1, off"
                     :: "v"(lds_off), "v"(gsrc) : "memory");
    }
    asm volatile("s_wait_asynccnt 0x0" ::: "memory");
    __syncthreads();

    const int lane = tid & 31;
    const int wave = tid >> 5;
    const int n0   = wave * 16;
    const int mr   = lane & 15;              // A: M index / B,D: N index within tile
    const int koff = (lane >> 4) << 1;       // 0 or 2 (K-pair select)
    const int ncol = n0 + mr;

    const float  nm   = norm_in[m0 + mr];    // norm_in scaling folded into A-fragment load
    const float* arow = &sA[mr * 132 + koff];
    const float* wcol = W + ncol;

    v8f acc = {};
#pragma unroll
    for (int k0 = 0; k0 < GNN_HID; k0 += 4) {
        float2 av = *(const float2*)(arow + k0);   // A[m][k0+koff], A[m][k0+koff+1]
        v2f a, b;
        a.x = av.x * nm;
        a.y = av.y * nm;
        b.x = wcol[(k0 + koff) * GNN_HID];          // B[k0+koff][n]
        b.y = wcol[(k0 + koff + 1) * GNN_HID];      // B[k0+koff+1][n]
        acc = __builtin_amdgcn_wmma_f32_16x16x4_f32(false, a, false, b,
                                                    (short)0, acc, false, false);
    }

    // ---- spill tile (+bias) to LDS: VGPR r -> row (r + 8*(lane>=16)), col ncol ----
    const float bv    = bias[ncol];
    const int   rbase = (lane >> 4) << 3;
#pragma unroll
    for (int r = 0; r < 8; ++r)
        sY[(rbase + r) * 132 + ncol] = acc[r] + bv;
    __syncthreads();

    // ---- LayerNorm + LeakyReLU + norm_out prescale: wave handles rows 2w, 2w+1 ----
#pragma unroll
    for (int rr = 0; rr < 2; ++rr) {
        const int row = wave * 2 + rr;
        float4 v = *(const float4*)(&sY[row * 132 + lane * 4]);
        float s  = v.x + v.y + v.z + v.w;
        float s2 = v.x * v.x + v.y * v.y + v.z * v.z + v.w * v.w;
#pragma unroll
        for (int off = 16; off > 0; off >>= 1) {
            s  += __shfl_xor(s,  off, 32);
            s2 += __shfl_xor(s2, off, 32);
        }
        float mu  = s * (1.0f / GNN_HID);
        float var = s2 * (1.0f / GNN_HID) - mu * mu;
        float rs  = rsqrtf(var + 1e-5f);
        float no  = norm_out[m0 + row];
        float4 gg = ((const float4*)g)[lane];
        float4 bb = ((const float4*)be)[lane];
        float4 o;
        float t;
        t = (v.x - mu) * rs * gg.x + bb.x; o.x = (t > 0.0f ? t : GNN_LEAKY * t) * no;
        t = (v.y - mu) * rs * gg.y + bb.y; o.y = (t > 0.0f ? t : GNN_LEAKY * t) * no;
        t = (v.z - mu) * rs * gg.z + bb.z; o.z = (t > 0.0f ? t : GNN_LEAKY * t) * no;
        t = (v.w - mu) * rs * gg.w + bb.w; o.w = (t > 0.0f ? t : GNN_LEAKY * t) * no;
        ((float4*)(h_out + (size_t)(m0 + row) * GNN_HID))[lane] = o;
    }
}

// ---------------- layer 3: out[n,0..1] = (agg*norm_in) @ W3 + b3 (wave per node) ---------
__global__ void gnn_out_kernel(const float* __restrict__ agg, const float* __restrict__ norm_in,
                               const float* __restrict__ W3, const float* __restrict__ b3,
                               float* __restrict__ out, int n) {
    int row  = (blockIdx.x * blockDim.x + threadIdx.x) >> 5;
    int lane = threadIdx.x & 31;
    if (row >= n) return;
    float4 v  = ((const float4*)(agg + (size_t)row * GNN_HID))[lane];
    // W3 is [128,2] row-major; lane covers k = 4*lane..4*lane+3  -> 8 consecutive floats
    float4 wA = ((const float4*)W3)[2 * lane];
    float4 wB = ((const float4*)W3)[2 * lane + 1];
    float s0 = v.x * wA.x + v.y * wA.z + v.z * wB.x + v.w * wB.z;
    float s1 = v.x * wA.y + v.y * wA.w + v.z * wB.y + v.w * wB.w;
#pragma unroll
    for (int off = 16; off > 0; off >>= 1) {
        s0 += __shfl_xor(s0, off, 32);
        s1 += __shfl_xor(s1, off, 32);
    }
    if (lane == 0) {
        float nm = norm_in[row];
        out[(size_t)row * 2 + 0] = s0 * nm + b3[0];
        out[(size_t)row * 2 + 1] = s1 * nm + b3[1];
    }
}

extern "C" void kernel_launch(void* const* d_in, const int* in_sizes, int n_in,
                              void* d_out, int out_size, void* d_ws, size_t ws_size,
                              hipStream_t stream) {
    const float* features = (const float*)d_in[0];
    const int*   src      = (const int*)d_in[1];
    const int*   dst      = (const int*)d_in[2];
    const float* W1 = (const float*)d_in[3];
    const float* b1 = (const float*)d_in[4];
    const float* W2 = (const float*)d_in[5];
    const float* b2 = (const float*)d_in[6];
    const float* W3 = (const float*)d_in[7];
    const float* b3 = (const float*)d_in[8];
    const float* g1  = (const float*)d_in[9];
    const float* be1 = (const float*)d_in[10];
    const float* g2  = (const float*)d_in[11];
    const float* be2 = (const float*)d_in[12];

    const int N = GNN_N_NODES;
    const int E = GNN_N_EDGES;

    // workspace layout (norm_out | norm_in | h | agg)  ~51.6 MB
    const size_t padN = ((size_t)N * 4 + 255) & ~(size_t)255;       // 200192 B
    const size_t NB   = (size_t)N * GNN_HID * 4;                    // 25.6 MB
    char* ws = (char*)d_ws;
    float* norm_out = (float*)(ws);
    float* norm_in  = (float*)(ws + padN);
    float* h        = (float*)(ws + 2 * padN);
    float* agg      = (float*)(ws + 2 * padN + NB);

    // degrees + norms (recomputed every call: deterministic)
    hipMemsetAsync(norm_out, 0, 2 * padN, stream);
    gnn_degree_kernel<<<(E + 255) / 256, 256, 0, stream>>>(src, dst, norm_out, norm_in, E);
    gnn_norm_kernel<<<(N + 255) / 256, 256, 0, stream>>>(norm_out, N);
    gnn_norm_kernel<<<(N + 255) / 256, 256, 0, stream>>>(norm_in, N);

    // h = features * norm_out
    gnn_scale_kernel<<<(N * GNN_HID) / 256, 256, 0, stream>>>(features, norm_out, h, N * GNN_HID);

    const int scatter_blocks = (E * 32) / 256;   // warp per edge
    const int rowwave_blocks = (N * 32) / 256;   // wave per node row
    const int gemm_blocks    = N / 16;           // 50000 divisible by 16

    // ---- layer 1 ----
    hipMemsetAsync(agg, 0, NB, stream);
    gnn_scatter_kernel<<<scatter_blocks, 256, 0, stream>>>((const float4*)h, src, dst, agg, E);
    gnn_gconv_wmma_ln_kernel<<<gemm_blocks, 256, 0, stream>>>(agg, norm_in, W1, b1,
                                                              g1, be1, norm_out, h);

    // ---- layer 2 ----
    hipMemsetAsync(agg, 0, NB, stream);
    gnn_scatter_kernel<<<scatter_blocks, 256, 0, stream>>>((const float4*)h, src, dst, agg, E);
    gnn_gconv_wmma_ln_kernel<<<gemm_blocks, 256, 0, stream>>>(agg, norm_in, W2, b2,
                                                              g2, be2, norm_out, h);

    // ---- layer 3 (out_dim = 2) ----
    hipMemsetAsync(agg, 0, NB, stream);
    gnn_scatter_kernel<<<scatter_blocks, 256, 0, stream>>>((const float4*)h, src, dst, agg, E);
    gnn_out_kernel<<<rowwave_blocks, 256, 0, stream>>>(agg, norm_in, W3, b3, (float*)d_out, N);
}